// ICUSTOMIntGELU_80006650789912
// MI455X (gfx1250) — compile-verified
//
#include <hip/hip_runtime.h>
#include <cmath>

// ---------------------------------------------------------------------------
// IntGELU forward (I-BERT style) for MI455X / gfx1250.
//
// Memory-bound elementwise op: 403 MB of HBM traffic => ~17.3 us floor at
// 23.3 TB/s. No matrix structure => WMMA inapplicable; optimization target is
// b128 non-temporal streaming + <~30 VALU instructions per element.
//
// Per element (wave32):
//   xi  = floor(x * (1/s))                       (2 VALU)
//   g   = clamp(fma(xi, gs, go), 0, 15)          (4 VALU)  segment guess
//   T[g], T[g+1] from LDS (ds_read2_b32)         (1 DS)
//   idx = g + (xi>=T[g+1]) - (xi<T[g])           (4 VALU)  exact searchsorted
//   (a,b,c) = LDS coeff table [idx]              (1 DS, ds_load_b128)
//   r   = fma(fma(a,xi,b),xi,c)                  (2 VALU)
//   y   = floor(r * (1/s)) * s_out               (3 VALU)
// ---------------------------------------------------------------------------

typedef float v4f __attribute__((ext_vector_type(4)));

#define TWO_N_F 1048576.0f      // 2^20 fixed-point scale
#define FLT_BIG 3.402823466e38f

struct FitConsts {
  float lo[15];    // segment lower bounds lo_f[1..15] (float32 linspace values)
  float coef[48];  // per segment: a (x^2), b (x), c  (highest power first)
};

// d_ws float layout:
//  [0..16]   T[0..16] integer thresholds with +-BIG sentinels
//  [32..95]  16 segments x (c0, c1, c2, pad)  -- 16B stride for ds_load_b128
//  [96..99]  rcp_s, s_out, g_scale, g_off

__global__ void igelu_setup(const float* __restrict__ sf, FitConsts fc,
                            float* __restrict__ ws, float* __restrict__ s_out_ptr,
                            const float* __restrict__ x, float* __restrict__ y,
                            int tail_start, int n) {
  if (threadIdx.x != 0 || blockIdx.x != 0) return;
  float s = sf[0];

  // Integer segment thresholds, exactly as reference: floor(lo_f / s) in fp32.
  ws[0] = -FLT_BIG;
  for (int i = 1; i <= 15; ++i) ws[i] = floorf(fc.lo[i - 1] / s);
  ws[16] = FLT_BIG;

  // Fixed-point integer coefficients: floor(coef * s^p * 2^20), p = 2,1,0.
  float s2 = s * s;
  for (int k = 0; k < 16; ++k) {
    ws[32 + 4 * k + 0] = floorf(fc.coef[3 * k + 0] * s2 * TWO_N_F);
    ws[32 + 4 * k + 1] = floorf(fc.coef[3 * k + 1] * s  * TWO_N_F);
    ws[32 + 4 * k + 2] = floorf(fc.coef[3 * k + 2] * TWO_N_F);
    ws[32 + 4 * k + 3] = 0.0f;
  }

  float rcp = 1.0f / s;
  float so  = s / TWO_N_F;
  float T1 = ws[1], T15 = ws[15];
  float gs = (T15 > T1) ? (14.0f / (T15 - T1)) : 0.0f;  // ~1/threshold spacing
  ws[96] = rcp;
  ws[97] = so;
  ws[98] = gs;
  ws[99] = 1.0f - T1 * gs;  // g ~ (xi - T1)*gs + 1
  *s_out_ptr = so;          // second tuple output: s_out

  // Scalar tail if n % 4 != 0 (not hit for the reference shape).
  for (int j = tail_start; j < n; ++j) {
    float xi = floorf(x[j] * rcp);
    int idx = 0;
    for (int i = 1; i <= 15; ++i) idx += (xi >= ws[i]) ? 1 : 0;
    float r = fmaf(fmaf(ws[32 + 4 * idx], xi, ws[32 + 4 * idx + 1]), xi,
                   ws[32 + 4 * idx + 2]);
    y[j] = floorf(r * rcp) * so;
  }
}

__global__ __launch_bounds__(256) void igelu_main(const float* __restrict__ x,
                                                  float* __restrict__ y,
                                                  const float* __restrict__ ws,
                                                  int n4) {
  __shared__ float sT[17];
  __shared__ float sS[4];
  __shared__ v4f   sC[16];

  int t = threadIdx.x;
  if (t < 17) sT[t] = ws[t];
  if (t < 4)  sS[t] = ws[96 + t];
  if (t < 64) ((float*)sC)[t] = ws[32 + t];
  __syncthreads();

  float rcp = sS[0], so = sS[1], gs = sS[2], go = sS[3];

  int i = blockIdx.x * 256 + t;
  if (i >= n4) return;

  v4f v = __builtin_nontemporal_load((const v4f*)x + i);
  v4f o;
#pragma unroll
  for (int k = 0; k < 4; ++k) {
    float xv = v[k];
    float xi = floorf(xv * rcp);                 // quantized integer (as float)
    float gf = fmaf(xi, gs, go);                 // segment guess (+-1 accurate)
    gf = fminf(fmaxf(gf, 0.0f), 15.0f);
    int g = (int)gf;                             // trunc == floor (gf >= 0)
    float tg  = sT[g];                           // ds_read2_b32 pair
    float tg1 = sT[g + 1];
    int idx = g + (int)(xi >= tg1) - (int)(xi < tg);  // exact searchsorted
    v4f c = sC[idx];                             // ds_load_b128 coeff gather
    float r = fmaf(fmaf(c[0], xi, c[1]), xi, c[2]);   // Horner, fixed point
    o[k] = floorf(r * rcp) * so;                 // requantize to s_out grid
  }
  __builtin_nontemporal_store(o, (v4f*)y + i);
}

// ---------------------------------------------------------------------------
// Host-side reconstruction of the data-independent piecewise quadratic fit of
// exact GELU on [-5,5], 16 segments, 10000-point grid (double-precision
// closed-form least squares; matches numpy's fp32 polyfit far below the
// scheme's own +-1 integer-coefficient flooring noise).
// ---------------------------------------------------------------------------
static void solve3(double A[3][3], double b[3], double xo[3]) {
  for (int col = 0; col < 3; ++col) {
    int piv = col;
    for (int r = col + 1; r < 3; ++r)
      if (fabs(A[r][col]) > fabs(A[piv][col])) piv = r;
    if (piv != col) {
      for (int k = 0; k < 3; ++k) { double tmp = A[col][k]; A[col][k] = A[piv][k]; A[piv][k] = tmp; }
      double tb = b[col]; b[col] = b[piv]; b[piv] = tb;
    }
    for (int r = col + 1; r < 3; ++r) {
      double f = A[r][col] / A[col][col];
      for (int k = col; k < 3; ++k) A[r][k] -= f * A[col][k];
      b[r] -= f * b[col];
    }
  }
  for (int r = 2; r >= 0; --r) {
    double v = b[r];
    for (int k = r + 1; k < 3; ++k) v -= A[r][k] * xo[k];
    xo[r] = v / A[r][r];
  }
}

static void build_fit(FitConsts& fc) {
  const int NP = 10000;
  float bff[17];
  for (int k = 0; k <= 16; ++k) bff[k] = (float)(-5.0 + 10.0 * (double)k / 16.0);

  for (int seg = 0; seg < 16; ++seg) {
    double s0 = 0, s1 = 0, s2 = 0, s3 = 0, s4 = 0, b0 = 0, b1 = 0, b2 = 0;
    for (int i = 0; i < NP; ++i) {
      float xf = (float)(-5.0 + 10.0 * (double)i / 9999.0);  // fp32 linspace
      if (!(xf >= bff[seg] && xf <= bff[seg + 1])) continue; // inclusive bounds
      double X = (double)xf;
      // exact GELU, rounded to fp32 like the reference's ys
      double Y = (double)(float)(0.5 * X * (1.0 + erf(X * 0.7071067811865476)));
      double X2 = X * X;
      s0 += 1.0; s1 += X; s2 += X2; s3 += X2 * X; s4 += X2 * X2;
      b0 += Y;  b1 += X * Y; b2 += X2 * Y;
    }
    double M[3][3] = {{s4, s3, s2}, {s3, s2, s1}, {s2, s1, s0}};
    double rhs[3] = {b2, b1, b0};
    double c[3];
    solve3(M, rhs, c);
    fc.coef[3 * seg + 0] = (float)c[0];
    fc.coef[3 * seg + 1] = (float)c[1];
    fc.coef[3 * seg + 2] = (float)c[2];
  }
  for (int i = 1; i <= 15; ++i) fc.lo[i - 1] = bff[i];
}

extern "C" void kernel_launch(void* const* d_in, const int* in_sizes, int n_in,
                              void* d_out, int out_size, void* d_ws, size_t ws_size,
                              hipStream_t stream) {
  (void)n_in; (void)out_size; (void)ws_size;
  const float* x  = (const float*)d_in[0];
  const float* sf = (const float*)d_in[1];
  float* y  = (float*)d_out;
  float* ws = (float*)d_ws;
  int n  = in_sizes[0];
  int n4 = n / 4;

  FitConsts fc;
  build_fit(fc);  // deterministic, data-independent; host-side, capture-safe

  // s-dependent constants + s_out output + scalar tail (device: s lives in HBM)
  igelu_setup<<<1, 32, 0, stream>>>(sf, fc, ws, y + n, x, y, n4 * 4, n);

  if (n4 > 0) {
    int blocks = (n4 + 255) / 256;
    igelu_main<<<blocks, 256, 0, stream>>>(x, y, ws, n4);
  }
}